// AspectAwareDynamicGCN_88965952569837
// MI455X (gfx1250) — compile-verified
//
#include <hip/hip_runtime.h>
#include <hip/hip_bf16.h>

// B=8, L=256, D=128, H=6, E=32
#define B_ 8
#define L_ 256
#define D_ 128
#define H_ 6
#define E_ 32

typedef __attribute__((ext_vector_type(2))) float v2f;
typedef __attribute__((ext_vector_type(8))) float v8f;

// ---------------------------------------------------------------------------
// fp32 WMMA 16x16x4 (CDNA5): D = A(16x4) * B(4x16) + C(16x16), full precision.
// Fragment layout (ISA 7.12.2, wave32):
//   A: lane l holds A[l%16][(l/16)*2 + v], v=0..1
//   B: lane l holds B[(l/16)*2 + v][l%16]
//   C/D: lane l, reg r holds D[(l/16)*8 + r][l%16]
// ---------------------------------------------------------------------------
__device__ __forceinline__ v8f wmma_f32_16x16x4(v2f a, v2f b, v8f c) {
#if __has_builtin(__builtin_amdgcn_wmma_f32_16x16x4_f32)
  return __builtin_amdgcn_wmma_f32_16x16x4_f32(
      /*neg_a=*/false, a, /*neg_b=*/false, b,
      /*c_mod=*/(short)0, c, /*reuse_a=*/false, /*reuse_b=*/false);
#else
  // Cross-lane emulation with the same fragment convention (fallback only).
  const int lane = (int)(threadIdx.x & 31u);
  const int half = lane >> 4;
  const int col  = lane & 15;
#pragma unroll
  for (int r = 0; r < 8; ++r) {
    const int row = half * 8 + r;
    float s = c[r];
#pragma unroll
    for (int k = 0; k < 4; ++k) {
      float av = __shfl((k & 1) ? a.y : a.x, row + ((k >> 1) << 4), 32);
      float bv = __shfl((k & 1) ? b.y : b.x, col + ((k >> 1) << 4), 32);
      s += av * bv;
    }
    c[r] = s;
  }
  return c;
#endif
}

// ---------------------------------------------------------------------------
// Kernel 1: per (b, 16-row i-tile):
//   adj_avg tile -> t = adj_avg @ x[b] -> gcn = relu(t @ Ww^T + Wb)
//   out = gcn @ Wxx^T + Wxxb ; n1 = gcn @ Wn1^T ; n2 = gcn @ Wn2^T
// Block: 128 threads (4 waves). Wave w owns N-tiles {32w..32w+15, 32w+16..32w+31}.
// ---------------------------------------------------------------------------
__global__ __launch_bounds__(128) void k1_gcn(
    const float* __restrict__ x, const float* __restrict__ wadj,
    const float* __restrict__ Ww, const float* __restrict__ Wb,
    const float* __restrict__ Wxw, const float* __restrict__ Wxxw,
    const float* __restrict__ Wxxb,
    float* __restrict__ out, float* __restrict__ n1ws, float* __restrict__ n2ws)
{
  __shared__ __align__(16) float lds_adj[16 * 256];  // 16 KB: adj_avg tile (16 x L)
  __shared__ __align__(16) float lds_x[64 * 128];    // 32 KB: K-chunk of x[b]
  __shared__ __align__(16) float lds_t[16 * 128];    //  8 KB: t tile, then gcn tile

  const int itile = blockIdx.x;        // 0..15
  const int b     = blockIdx.y;        // 0..7
  const int tid   = (int)threadIdx.x;
  const int lane  = tid & 31;
  const int wave  = tid >> 5;
  const int half  = lane >> 4;         // 0/1
  const int l16   = lane & 15;
  const int n0    = wave * 32;

  // ---- Step A: adj_avg tile (mean over H) into LDS -------------------------
  {
    const float inv_h = 1.0f / (float)H_;
    const size_t wb = (size_t)b * H_ * L_ * L_;
    for (int idx = tid; idx < 16 * 256; idx += 128) {
      const int r = idx >> 8, j = idx & 255;
      const size_t base = wb + (size_t)(itile * 16 + r) * L_ + j;
      float s = 0.f;
#pragma unroll
      for (int h = 0; h < H_; ++h) s += wadj[base + (size_t)h * (L_ * L_)];
      lds_adj[idx] = s * inv_h;
    }
  }

  // ---- GEMM1: t(16x128) = adj_tile(16x256) @ x[b](256x128) -----------------
  v8f acc0 = {}; v8f acc1 = {};
  const float* xb = x + (size_t)b * L_ * D_;
  for (int kc = 0; kc < 4; ++kc) {          // K chunks of 64
    __syncthreads();
    for (int idx = tid; idx < (64 * 128) / 4; idx += 128)
      ((float4*)lds_x)[idx] = ((const float4*)(xb + kc * 64 * 128))[idx];
    __syncthreads();
#pragma unroll
    for (int ks = 0; ks < 16; ++ks) {       // 16 K-steps of 4
      const int k0 = ks * 4;
      v2f a, b0, b1;
      a.x  = lds_adj[l16 * 256 + kc * 64 + k0 + half * 2 + 0];
      a.y  = lds_adj[l16 * 256 + kc * 64 + k0 + half * 2 + 1];
      b0.x = lds_x[(k0 + half * 2 + 0) * 128 + n0 + l16];
      b0.y = lds_x[(k0 + half * 2 + 1) * 128 + n0 + l16];
      b1.x = lds_x[(k0 + half * 2 + 0) * 128 + n0 + 16 + l16];
      b1.y = lds_x[(k0 + half * 2 + 1) * 128 + n0 + 16 + l16];
      acc0 = wmma_f32_16x16x4(a, b0, acc0);
      acc1 = wmma_f32_16x16x4(a, b1, acc1);
    }
  }
  __syncthreads();
#pragma unroll
  for (int r = 0; r < 8; ++r) {
    const int row = half * 8 + r;
    lds_t[row * 128 + n0 + l16]      = acc0[r];
    lds_t[row * 128 + n0 + 16 + l16] = acc1[r];
  }
  __syncthreads();

  // ---- GEMM2: gcn = relu(t @ Ww^T + Wb) ------------------------------------
  v8f g0 = {}; v8f g1 = {};
#pragma unroll 8
  for (int ks = 0; ks < 32; ++ks) {
    const int k0 = ks * 4;
    v2f a, b0, b1;
    a.x  = lds_t[l16 * 128 + k0 + half * 2 + 0];
    a.y  = lds_t[l16 * 128 + k0 + half * 2 + 1];
    b0.x = Ww[(size_t)(n0 + l16) * 128 + k0 + half * 2 + 0];       // Ww^T[k][n]
    b0.y = Ww[(size_t)(n0 + l16) * 128 + k0 + half * 2 + 1];
    b1.x = Ww[(size_t)(n0 + 16 + l16) * 128 + k0 + half * 2 + 0];
    b1.y = Ww[(size_t)(n0 + 16 + l16) * 128 + k0 + half * 2 + 1];
    g0 = wmma_f32_16x16x4(a, b0, g0);
    g1 = wmma_f32_16x16x4(a, b1, g1);
  }
  {
    const float bv0 = Wb[n0 + l16], bv1 = Wb[n0 + 16 + l16];
#pragma unroll
    for (int r = 0; r < 8; ++r) {
      g0[r] = fmaxf(g0[r] + bv0, 0.f);
      g1[r] = fmaxf(g1[r] + bv1, 0.f);
    }
  }
  __syncthreads();                            // all waves done reading t
#pragma unroll
  for (int r = 0; r < 8; ++r) {
    const int row = half * 8 + r;
    lds_t[row * 128 + n0 + l16]      = g0[r];  // lds_t now holds gcn tile
    lds_t[row * 128 + n0 + 16 + l16] = g1[r];
  }
  __syncthreads();

  // ---- GEMM3: out = gcn @ Wxx^T + Wxxb -------------------------------------
  v8f o0 = {}; v8f o1 = {};
#pragma unroll 8
  for (int ks = 0; ks < 32; ++ks) {
    const int k0 = ks * 4;
    v2f a, b0, b1;
    a.x  = lds_t[l16 * 128 + k0 + half * 2 + 0];
    a.y  = lds_t[l16 * 128 + k0 + half * 2 + 1];
    b0.x = Wxxw[(size_t)(n0 + l16) * 128 + k0 + half * 2 + 0];
    b0.y = Wxxw[(size_t)(n0 + l16) * 128 + k0 + half * 2 + 1];
    b1.x = Wxxw[(size_t)(n0 + 16 + l16) * 128 + k0 + half * 2 + 0];
    b1.y = Wxxw[(size_t)(n0 + 16 + l16) * 128 + k0 + half * 2 + 1];
    o0 = wmma_f32_16x16x4(a, b0, o0);
    o1 = wmma_f32_16x16x4(a, b1, o1);
  }
  {
    const float cv0 = Wxxb[n0 + l16], cv1 = Wxxb[n0 + 16 + l16];
    const size_t obase = (size_t)b * L_ * D_ + (size_t)(itile * 16) * D_;
#pragma unroll
    for (int r = 0; r < 8; ++r) {
      const int row = half * 8 + r;
      out[obase + (size_t)row * D_ + n0 + l16]      = o0[r] + cv0;
      out[obase + (size_t)row * D_ + n0 + 16 + l16] = o1[r] + cv1;
    }
  }

  // ---- n1/n2: gcn @ Wn1^T, gcn @ Wn2^T (N=6: VALU) -------------------------
  if (tid < 16 * H_) {
    const int il = tid / H_, k = tid % H_;
    const float* grow = lds_t + il * 128;
    const float* w1 = Wxw + (size_t)k * (H_ + 2 * D_ + E_) + H_;        // Wn1[k][:]
    const float* w2 = w1 + D_;                                          // Wn2[k][:]
    float s1 = 0.f, s2 = 0.f;
#pragma unroll 8
    for (int d = 0; d < D_; ++d) { s1 += grow[d] * w1[d]; s2 += grow[d] * w2[d]; }
    const int i = itile * 16 + il;
    n1ws[((size_t)b * L_ + i) * H_ + k] = s1;
    n2ws[((size_t)b * L_ + i) * H_ + k] = s2;
  }
}

// ---------------------------------------------------------------------------
// Kernel 2: streaming new_adj[b,k,i,j] =
//   sum_h wadj[b,h,i,j]*Wa[k,h] + n1[b,i,k] + n2[b,j,k]
//   + sum_e e[b,i,j,e]*We[k,e] + Wxb[k]
// One block per (b,i); thread = j. Bandwidth-bound: 152 B in / 24 B out, 228 FMA.
// ---------------------------------------------------------------------------
__global__ __launch_bounds__(256) void k2_adj(
    const float* __restrict__ wadj, const float* __restrict__ e,
    const float* __restrict__ Wxw, const float* __restrict__ Wxb,
    const float* __restrict__ n1ws, const float* __restrict__ n2ws,
    float* __restrict__ outadj)
{
  __shared__ float lWa[H_ * H_];
  __shared__ float lWe[H_ * E_];
  __shared__ float lb[H_];
  __shared__ float ln1[H_];

  const int i = blockIdx.x, b = blockIdx.y, j = (int)threadIdx.x;
  const int WXROW = H_ + 2 * D_ + E_;  // 294
  if (j < H_ * H_) lWa[j] = Wxw[(j / H_) * WXROW + (j % H_)];
  if (j < H_ * E_) lWe[j] = Wxw[(j / E_) * WXROW + (H_ + 2 * D_) + (j % E_)];
  if (j < H_) { lb[j] = Wxb[j]; ln1[j] = n1ws[((size_t)b * L_ + i) * H_ + j]; }
  __syncthreads();

  float adjv[H_];
  const size_t wbase = (size_t)b * H_ * L_ * L_ + (size_t)i * L_ + j;
#pragma unroll
  for (int h = 0; h < H_; ++h) adjv[h] = wadj[wbase + (size_t)h * (L_ * L_)];

  float ev[E_];
  const float4* e4 = (const float4*)(e + (((size_t)b * L_ + i) * L_ + j) * E_);
#pragma unroll
  for (int q = 0; q < E_ / 4; ++q) {
    float4 v = e4[q];
    ev[4 * q + 0] = v.x; ev[4 * q + 1] = v.y; ev[4 * q + 2] = v.z; ev[4 * q + 3] = v.w;
  }

  float acc[H_];
#pragma unroll
  for (int k = 0; k < H_; ++k) {
    float s = lb[k] + ln1[k] + n2ws[((size_t)b * L_ + j) * H_ + k];
#pragma unroll
    for (int h = 0; h < H_; ++h) s += adjv[h] * lWa[k * H_ + h];
#pragma unroll
    for (int t = 0; t < E_; ++t) s += ev[t] * lWe[k * E_ + t];
    acc[k] = s;
  }

  const size_t ob = (size_t)b * H_ * L_ * L_ + (size_t)i * L_ + j;
#pragma unroll
  for (int k = 0; k < H_; ++k) outadj[ob + (size_t)k * (L_ * L_)] = acc[k];
}

// ---------------------------------------------------------------------------
extern "C" void kernel_launch(void* const* d_in, const int* in_sizes, int n_in,
                              void* d_out, int out_size, void* d_ws, size_t ws_size,
                              hipStream_t stream) {
  const float* x    = (const float*)d_in[0];
  const float* wadj = (const float*)d_in[1];
  const float* e    = (const float*)d_in[2];
  const float* Ww   = (const float*)d_in[3];
  const float* Wb   = (const float*)d_in[4];
  const float* Wxw  = (const float*)d_in[5];
  const float* Wxb  = (const float*)d_in[6];
  const float* Wxxw = (const float*)d_in[7];
  const float* Wxxb = (const float*)d_in[8];

  float* out    = (float*)d_out;              // (B,L,D) = 262144 floats, first
  float* outadj = out + (size_t)B_ * L_ * D_; // (B,H,L,L) = 3145728 floats

  float* n1ws = (float*)d_ws;                 // B*L*H floats
  float* n2ws = n1ws + (size_t)B_ * L_ * H_;  // B*L*H floats (total 96 KB)

  k1_gcn<<<dim3(L_ / 16, B_), 128, 0, stream>>>(x, wadj, Ww, Wb, Wxw, Wxxw, Wxxb,
                                                out, n1ws, n2ws);
  k2_adj<<<dim3(L_, B_), 256, 0, stream>>>(wadj, e, Wxw, Wxb, n1ws, n2ws, outadj);
}